// FClip_22514218565760
// MI455X (gfx1250) — compile-verified
//
#include <hip/hip_runtime.h>
#include <math.h>

typedef float v2f __attribute__((ext_vector_type(2)));
typedef float v8f __attribute__((ext_vector_type(8)));

#define WW 2048
#define HH 2048
#define HW (WW * HH)
#define TOPK 1000
#define KPAD 1024
#define NBINS 8192          // jloc in (0, 1.0]; float-bits>>17 <= 8128
#define CAP 8192
#define SOFT_F 0.8f
#define RADIUS_F 64.0f
#define STRUCT_T 2.0f
#define PI_F 3.14159265358979323846f

#define NMS_BLOCKS 512
#define PIX_PER_BLK (HW / NMS_BLOCKS)   // 8192 contiguous pixels (4 rows) per block

__device__ __forceinline__ float sigmoidf_(float x) { return 1.0f / (1.0f + __expf(-x)); }

// ---------------- init ----------------
__global__ void k_init(unsigned* __restrict__ hist) {
    int i = blockIdx.x * blockDim.x + threadIdx.x;
    if (i < NBINS + 16) hist[i] = 0u;
}

// ---------------- pass 1 (fused): t = h1-h0 stencil, 3x3 peak NMS, jloc, LDS histogram ----------------
__global__ __launch_bounds__(256) void k_nms_hist(const float* __restrict__ hm,
                                                  float* __restrict__ jloc,
                                                  unsigned* __restrict__ hist) {
    __shared__ unsigned lh[NBINS];      // 32 KB
    for (int b = threadIdx.x; b < NBINS; b += blockDim.x) lh[b] = 0u;
    __syncthreads();

    const float* __restrict__ h0 = hm;
    const float* __restrict__ h1 = hm + HW;
    int base = blockIdx.x * PIX_PER_BLK;
    for (int p = threadIdx.x; p < PIX_PER_BLK; p += blockDim.x) {
        int i = base + p;
        int y = i >> 11, x = i & (WW - 1);
        float c = h1[i] - h0[i];
        float m = c;
        #pragma unroll
        for (int dy = -1; dy <= 1; ++dy) {
            int yy = y + dy;
            if (yy < 0 || yy >= HH) continue;
            #pragma unroll
            for (int dx = -1; dx <= 1; ++dx) {
                int xx = x + dx;
                if (xx < 0 || xx >= WW) continue;
                int j = yy * WW + xx;
                m = fmaxf(m, h1[j] - h0[j]);
            }
        }
        float s = sigmoidf_(c);
        float v = (c == m) ? s : s * SOFT_F;  // sigmoid monotone: t==max(t) <=> softmax==pooled
        jloc[i] = v;
        atomicAdd(&lh[__float_as_uint(v) >> 17], 1u);
    }
    __syncthreads();
    for (int b = threadIdx.x; b < NBINS; b += blockDim.x) {
        unsigned c = lh[b];
        if (c) atomicAdd(&hist[b], c);
    }
}

// ---------------- pass 2: find threshold bin (two-level scan) ----------------
__global__ __launch_bounds__(256) void k_thresh(const unsigned* __restrict__ hist,
                                                unsigned* __restrict__ meta) {
    __shared__ unsigned csum[256];
    int tid = threadIdx.x;
    unsigned s = 0;
    for (int b = tid * 32; b < tid * 32 + 32; ++b) s += hist[b];
    csum[tid] = s;
    __syncthreads();
    if (tid == 0) {
        unsigned cum = 0; int C = 0;
        for (int c = 255; c >= 0; --c) {
            if (cum + csum[c] >= TOPK) { C = c; break; }
            cum += csum[c];
        }
        int T = 0;
        for (int b = C * 32 + 31; b >= C * 32; --b) {
            cum += hist[b];
            if (cum >= TOPK) { T = b; break; }
        }
        meta[0] = (unsigned)T;
    }
}

// ---------------- pass 3: compact candidates >= threshold bin ----------------
__global__ void k_compact(const float* __restrict__ jloc, unsigned* __restrict__ meta,
                          uint2* __restrict__ cand) {
    int i = blockIdx.x * blockDim.x + threadIdx.x;
    if (i >= HW) return;
    float v = jloc[i];
    unsigned bits = __float_as_uint(v);
    if ((bits >> 17) >= meta[0]) {
        unsigned p = atomicAdd(&meta[1], 1u);
        if (p < CAP) cand[p] = make_uint2(~(unsigned)i, bits);  // lo = ~idx, hi = value bits
    }
}

// ---------------- pass 4: one-block bitonic sort + build line geometry ----------------
__global__ __launch_bounds__(1024) void k_sort_lines(
        const float* __restrict__ hm, const unsigned* __restrict__ meta,
        const uint2* __restrict__ cand, float* __restrict__ scores,
        float* __restrict__ L, float* __restrict__ Lsw, float* __restrict__ nrm) {
    __shared__ unsigned long long keys[CAP];   // 64 KB LDS
    int tid = threadIdx.x;
    int n = (int)meta[1]; if (n > CAP) n = CAP;
    for (int i = tid; i < CAP; i += blockDim.x) {
        keys[i] = (i < n) ? ((((unsigned long long)cand[i].y) << 32) | (unsigned long long)cand[i].x)
                          : 0ull;
    }
    __syncthreads();
    // bitonic sort, descending by key = (valbits, ~idx)  => value desc, index asc on ties
    for (int k = 2; k <= CAP; k <<= 1) {
        for (int j = k >> 1; j > 0; j >>= 1) {
            for (int i = tid; i < CAP; i += blockDim.x) {
                int ixj = i ^ j;
                if (ixj > i) {
                    unsigned long long a = keys[i], b = keys[ixj];
                    bool up = ((i & k) == 0);
                    if (up ? (a < b) : (a > b)) { keys[i] = b; keys[ixj] = a; }
                }
            }
            __syncthreads();
        }
    }
    if (tid < KPAD) {
        float l0 = 0.f, l1 = 0.f, l2 = 0.f, l3 = 0.f, nn = 0.f;
        if (tid < TOPK) {
            unsigned long long kk = keys[tid];
            int idx = (int)(~(unsigned)(kk & 0xFFFFFFFFull));
            scores[tid] = __uint_as_float((unsigned)(kk >> 32));
            int yy = idx >> 11, xx = idx & (WW - 1);
            float ox = sigmoidf_(hm[3 * HW + idx]);           // joff0 (x)
            float oy = sigmoidf_(hm[2 * HW + idx]);           // joff1 (y)
            float X = (float)xx + ox, Y = (float)yy + oy;
            float r = sigmoidf_(hm[4 * HW + idx]) * RADIUS_F;
            float a = sigmoidf_(hm[5 * HW + idx]) * PI_F;
            float dx = cosf(a) * r;
            float dy = -fabsf(sinf(a)) * r;
            l0 = X + dx; l1 = Y + dy; l2 = X - dx; l3 = Y - dy;
            nn = l0 * l0 + l1 * l1 + l2 * l2 + l3 * l3;
        }
        L[tid * 4 + 0] = l0;  L[tid * 4 + 1] = l1;  L[tid * 4 + 2] = l2;  L[tid * 4 + 3] = l3;
        Lsw[tid * 4 + 0] = l2; Lsw[tid * 4 + 1] = l3; Lsw[tid * 4 + 2] = l0; Lsw[tid * 4 + 3] = l1;
        nrm[tid] = nn;
    }
}

// ---------------- pass 5: WMMA pairwise-distance adjacency ----------------
// grid = 64 waves (blocks of 32); block b owns rows [16b, 16b+16).
// G1 = L.L^T, G2 = Lsw.L^T via V_WMMA_F32_16X16X4_F32; d = n_i + n_j - 2*max(G1,G2).
__global__ __launch_bounds__(32) void k_adj(const float* __restrict__ L,
                                            const float* __restrict__ Lsw,
                                            const float* __restrict__ nrm,
                                            unsigned* __restrict__ adj) {
    const v2f* __restrict__ Lv   = (const v2f*)L;
    const v2f* __restrict__ Lswv = (const v2f*)Lsw;
    int lane = threadIdx.x;
    int m15 = lane & 15;
    int hi  = lane >> 4;            // 0: lanes 0-15 (K=0,1 / M=v), 1: lanes 16-31 (K=2,3 / M=v+8)
    int itile = blockIdx.x;
    int arow = itile * 16 + m15;

    v2f A   = Lv[arow * 2 + hi];
    v2f Asw = Lswv[arow * 2 + hi];

    float nv[8];
    #pragma unroll
    for (int v = 0; v < 8; ++v) nv[v] = nrm[itile * 16 + hi * 8 + v];

    for (int jp = 0; jp < 32; ++jp) {
        unsigned word = 0;
        #pragma unroll
        for (int e = 0; e < 2; ++e) {
            int jt = jp * 2 + e;
            v2f B = Lv[(jt * 16 + m15) * 2 + hi];
            v8f z = {};
            v8f g1 = __builtin_amdgcn_wmma_f32_16x16x4_f32(false, A,   false, B, (short)0, z, false, false);
            v8f g2 = __builtin_amdgcn_wmma_f32_16x16x4_f32(false, Asw, false, B, (short)0, z, false, false);

            int col = jt * 16 + m15;
            float nj = nrm[col];
            unsigned mball[8];
            #pragma unroll
            for (int v = 0; v < 8; ++v) {
                float g = fmaxf(g1[v], g2[v]);
                int row = itile * 16 + hi * 8 + v;
                float d = nv[v] + nj - 2.0f * g;
                bool pred = (d <= STRUCT_T) && (row != col) && (row < TOPK) && (col < TOPK);
                mball[v] = __builtin_amdgcn_ballot_w32(pred);
            }
            // lane r (<16) picks halfword for row r: r<8 -> low16 of mball[r], else high16 of mball[r-8]
            unsigned hsel = 0;
            #pragma unroll
            for (int v = 0; v < 8; ++v)
                if ((lane & 7) == v) hsel = (lane < 8) ? (mball[v] & 0xFFFFu) : (mball[v] >> 16);
            word |= (e == 0) ? hsel : (hsel << 16);
        }
        if (lane < 16) adj[(itile * 16 + lane) * 32 + jp] = word;
    }
}

// ---------------- pass 6: sequential structural-NMS (LDS-staged) + final write ----------------
__global__ __launch_bounds__(1024) void k_final(const unsigned* __restrict__ adj,
                                                const float* __restrict__ L,
                                                const float* __restrict__ scores,
                                                float* __restrict__ out) {
    __shared__ unsigned sadj[KPAD * 32];  // 128 KB: full adjacency bitmask (LDS = 320 KB/WGP)
    __shared__ unsigned sdrop[32];
    int tid = threadIdx.x;
    for (int w = tid; w < KPAD * 32; w += blockDim.x) sadj[w] = adj[w];
    __syncthreads();

    if (tid < 32) {
        unsigned dropw = sadj[tid];          // drop = adj[0]
        for (int i = 1; i < TOPK - 2; ++i) { // fori_loop(1, K-2)
            unsigned wsel = (unsigned)__builtin_amdgcn_readlane((int)dropw, i >> 5);
            if (((wsel >> (i & 31)) & 1u) == 0u) {       // if !drop[i]
                unsigned a = sadj[i * 32 + tid];
                int lo = tid * 32;
                unsigned mask;
                if (i >= lo + 31)      mask = 0u;
                else if (i < lo)       mask = 0xFFFFFFFFu;
                else                   mask = 0xFFFFFFFFu << ((i - lo) + 1);  // iota > i
                dropw |= a & mask;
            }
        }
        sdrop[tid] = dropw;
    }
    __syncthreads();
    if (tid < TOPK) {
        bool keep = (((sdrop[tid >> 5] >> (tid & 31)) & 1u) == 0u);
        float f = keep ? 1.0f : 0.0f;
        out[tid * 4 + 0] = L[tid * 4 + 0] * f;
        out[tid * 4 + 1] = L[tid * 4 + 1] * f;
        out[tid * 4 + 2] = L[tid * 4 + 2] * f;
        out[tid * 4 + 3] = L[tid * 4 + 3] * f;
        out[4 * TOPK + tid] = scores[tid] * f;
    }
}

extern "C" void kernel_launch(void* const* d_in, const int* in_sizes, int n_in,
                              void* d_out, int out_size, void* d_ws, size_t ws_size,
                              hipStream_t stream) {
    const float* hm = (const float*)d_in[0];    // [1,6,2048,2048] f32
    float* out = (float*)d_out;                 // lines[1000*4] ++ scores[1000]

    // workspace layout
    float*    jloc  = (float*)d_ws;             // 4M f32
    unsigned* hist  = (unsigned*)(jloc + HW);   // 8192 u32
    unsigned* meta  = hist + NBINS;             // [0]=threshold bin, [1]=candidate count
    uint2*    cand  = (uint2*)(meta + 16);      // 8192 x (lo=~idx, hi=valbits)
    float*    scores= (float*)(cand + CAP);     // 1024 f32
    float*    L     = scores + KPAD;            // 1024x4 f32
    float*    Lsw   = L + KPAD * 4;             // 1024x4 f32
    float*    nrm   = Lsw + KPAD * 4;           // 1024 f32
    unsigned* adj   = (unsigned*)(nrm + KPAD);  // 1024x32 u32 bitmask

    k_init<<<(NBINS + 16 + 255) / 256, 256, 0, stream>>>(hist);
    k_nms_hist<<<NMS_BLOCKS, 256, 0, stream>>>(hm, jloc, hist);
    k_thresh<<<1, 256, 0, stream>>>(hist, meta);
    k_compact<<<HW / 256, 256, 0, stream>>>(jloc, meta, cand);
    k_sort_lines<<<1, 1024, 0, stream>>>(hm, meta, cand, scores, L, Lsw, nrm);
    k_adj<<<64, 32, 0, stream>>>(L, Lsw, nrm, adj);
    k_final<<<1, 1024, 0, stream>>>(adj, L, scores, out);
}